// GraphSAGE_37426345017677
// MI455X (gfx1250) — compile-verified
//
#include <hip/hip_runtime.h>

typedef __attribute__((ext_vector_type(2))) float v2f;
typedef __attribute__((ext_vector_type(8))) float v8f;

#define D_FEAT 64
#define H1 16

// ---------------------------------------------------------------------------
// Layer-1 node transform via V_WMMA_F32_16X16X4_F32 (full fp32 precision).
// One wave handles one 16-node tile: computes
//   y1 = x_tile @ Wl1^T          (aggregated over edges later)
//   z1 = x_tile @ Wr1^T + b1     (self term)
// A fragment (16x4 f32, 2 VGPRs): lane m(0..15) holds K=koff..koff+1,
// koff = 2*(lane>=16)  -> contiguous float2 load from x row.
// B fragment (4x16 f32, 2 VGPRs): lane n(0..15) holds B[koff..koff+1][n]
//   = W[n][koff..koff+1]          -> contiguous float2 load from weight row.
// C/D (16x16 f32, 8 VGPRs): VGPR r -> M=r (lanes 0..15) / M=r+8 (lanes 16..31).
// ---------------------------------------------------------------------------
__global__ __launch_bounds__(256) void sage_l1_transform(
    const float* __restrict__ x, const float* __restrict__ Wl1,
    const float* __restrict__ Wr1, const float* __restrict__ b1,
    float* __restrict__ y1, float* __restrict__ z1,
    int n_nodes, int n_tiles) {
  const int lane = threadIdx.x & 31;
  const int wave = threadIdx.x >> 5;
  const int tile = blockIdx.x * 8 + wave;
  if (tile >= n_tiles) return;           // wave-uniform exit (EXEC stays all-1s)

  const int mrow = lane & 15;            // A row index / B column index
  const int koff = (lane >> 4) << 1;     // 0 or 2

  int row = tile * 16 + mrow;
  if (row >= n_nodes) row = n_nodes - 1; // clamp loads; stores are guarded

  const float* ax = x   + (size_t)row  * D_FEAT + koff;
  const float* bl = Wl1 + (size_t)mrow * D_FEAT + koff;
  const float* br = Wr1 + (size_t)mrow * D_FEAT + koff;

  v8f accL = {};
  v8f accR = {};
#pragma unroll
  for (int kk = 0; kk < D_FEAT / 4; ++kk) {
    v2f a  = *(const v2f*)(ax + kk * 4);
    v2f wl = *(const v2f*)(bl + kk * 4);
    v2f wr = *(const v2f*)(br + kk * 4);
    // 8 args: (neg_a, A, neg_b, B, c_mod, C, reuse_a, reuse_b)
    accL = __builtin_amdgcn_wmma_f32_16x16x4_f32(
        false, a, false, wl, (short)0, accL, false, false);
    accR = __builtin_amdgcn_wmma_f32_16x16x4_f32(
        false, a, false, wr, (short)0, accR, false, false);
  }

  const int col   = lane & 15;
  const int mbase = (lane >> 4) << 3;    // 0 or 8
  const float bias = b1[col];
#pragma unroll
  for (int r = 0; r < 8; ++r) {
    int m = tile * 16 + mbase + r;
    if (m < n_nodes) {
      y1[(size_t)m * H1 + col] = accL[r];
      z1[(size_t)m * H1 + col] = accR[r] + bias;
    }
  }
}

// ---------------------------------------------------------------------------
// Edge scatter, layer 1: 16 lanes per edge move the 16-dim transformed
// feature; lane 0 also accumulates the destination in-degree.
// y1/agg1 are 6.4 MB each -> resident in the 192 MB L2, so the random
// gathers + fp32 atomics resolve at L2 bandwidth.
// ---------------------------------------------------------------------------
__global__ __launch_bounds__(256) void edge_agg16(
    const int* __restrict__ src, const int* __restrict__ dst,
    const float* __restrict__ y1, float* __restrict__ agg1,
    float* __restrict__ cnt, int n_edges) {
  long long t = (long long)blockIdx.x * blockDim.x + threadIdx.x;
  int e = (int)(t >> 4);
  int f = (int)(t & 15);
  if (e < n_edges) {
    int s = src[e];
    int d = dst[e];
    atomicAdd(&agg1[(size_t)d * H1 + f], y1[(size_t)s * H1 + f]);
    if (f == 0) atomicAdd(&cnt[d], 1.0f);
  }
}

// ---------------------------------------------------------------------------
// Per-node: mean-divide, add self term, ReLU, then both H2=1 dot products
// (hoisting Wl2 before the second aggregation -> scalar per edge later).
// ---------------------------------------------------------------------------
__global__ __launch_bounds__(256) void node_l2(
    const float* __restrict__ agg1, const float* __restrict__ z1,
    const float* __restrict__ cnt, const float* __restrict__ Wl2,
    const float* __restrict__ Wr2, const float* __restrict__ b2,
    float* __restrict__ s2, float* __restrict__ r2, int n_nodes) {
  int n = blockIdx.x * blockDim.x + threadIdx.x;
  if (n >= n_nodes) return;
  float inv = 1.0f / fmaxf(cnt[n], 1.0f);
  const float4* a4 = (const float4*)(agg1 + (size_t)n * H1);
  const float4* z4 = (const float4*)(z1 + (size_t)n * H1);
  float sl = 0.f, sr = 0.f;
#pragma unroll
  for (int q = 0; q < H1 / 4; ++q) {
    float4 a = a4[q];
    float4 z = z4[q];
    float h0 = fmaxf(a.x * inv + z.x, 0.0f);
    float h1 = fmaxf(a.y * inv + z.y, 0.0f);
    float h2 = fmaxf(a.z * inv + z.z, 0.0f);
    float h3 = fmaxf(a.w * inv + z.w, 0.0f);
    sl += h0 * Wl2[4 * q + 0] + h1 * Wl2[4 * q + 1] +
          h2 * Wl2[4 * q + 2] + h3 * Wl2[4 * q + 3];
    sr += h0 * Wr2[4 * q + 0] + h1 * Wr2[4 * q + 1] +
          h2 * Wr2[4 * q + 2] + h3 * Wr2[4 * q + 3];
  }
  s2[n] = sl;
  r2[n] = sr + b2[0];
}

// ---------------------------------------------------------------------------
// Edge scatter, layer 2: one scalar per edge (4 B instead of 64 B).
// ---------------------------------------------------------------------------
__global__ __launch_bounds__(256) void edge_agg1f(
    const int* __restrict__ src, const int* __restrict__ dst,
    const float* __restrict__ s2, float* __restrict__ agg2, int n_edges) {
  int e = blockIdx.x * blockDim.x + threadIdx.x;
  if (e < n_edges) atomicAdd(&agg2[dst[e]], s2[src[e]]);
}

__global__ __launch_bounds__(256) void finalize(
    const float* __restrict__ agg2, const float* __restrict__ cnt,
    const float* __restrict__ r2, float* __restrict__ out, int n_nodes) {
  int n = blockIdx.x * blockDim.x + threadIdx.x;
  if (n < n_nodes) out[n] = agg2[n] / fmaxf(cnt[n], 1.0f) + r2[n];
}

extern "C" void kernel_launch(void* const* d_in, const int* in_sizes, int n_in,
                              void* d_out, int out_size, void* d_ws, size_t ws_size,
                              hipStream_t stream) {
  const float* x   = (const float*)d_in[0];
  const int*   ei  = (const int*)d_in[1];
  const float* Wl1 = (const float*)d_in[2];
  const float* Wr1 = (const float*)d_in[3];
  const float* b1  = (const float*)d_in[4];
  const float* Wl2 = (const float*)d_in[5];
  const float* Wr2 = (const float*)d_in[6];
  const float* b2  = (const float*)d_in[7];
  float* out = (float*)d_out;

  const int n_nodes = in_sizes[0] / D_FEAT;
  const int n_edges = in_sizes[1] / 2;
  const int* src = ei;                 // edge_index[0]
  const int* dst = ei + n_edges;       // edge_index[1]

  // Workspace carve-up (floats): y1, z1, agg1 [N*16]; cnt, s2, r2, agg2 [N]
  float* ws   = (float*)d_ws;
  float* y1   = ws;
  float* z1   = y1  + (size_t)n_nodes * H1;
  float* agg1 = z1  + (size_t)n_nodes * H1;
  float* cnt  = agg1 + (size_t)n_nodes * H1;
  float* s2   = cnt + n_nodes;
  float* r2   = s2  + n_nodes;
  float* agg2 = r2  + n_nodes;

  hipMemsetAsync(agg1, 0, (size_t)n_nodes * H1 * sizeof(float), stream);
  hipMemsetAsync(cnt,  0, (size_t)n_nodes * sizeof(float), stream);
  hipMemsetAsync(agg2, 0, (size_t)n_nodes * sizeof(float), stream);

  const int n_tiles = (n_nodes + 15) / 16;
  const int blocks1 = (n_tiles + 7) / 8;   // 8 waves (tiles) per 256-thread block
  sage_l1_transform<<<blocks1, 256, 0, stream>>>(x, Wl1, Wr1, b1, y1, z1,
                                                 n_nodes, n_tiles);

  const long long t16 = (long long)n_edges * 16;
  const int blocks2 = (int)((t16 + 255) / 256);
  edge_agg16<<<blocks2, 256, 0, stream>>>(src, dst, y1, agg1, cnt, n_edges);

  const int blocksN = (n_nodes + 255) / 256;
  node_l2<<<blocksN, 256, 0, stream>>>(agg1, z1, cnt, Wl2, Wr2, b2, s2, r2,
                                       n_nodes);

  const int blocksE = (n_edges + 255) / 256;
  edge_agg1f<<<blocksE, 256, 0, stream>>>(src, dst, s2, agg2, n_edges);

  finalize<<<blocksN, 256, 0, stream>>>(agg2, cnt, r2, out, n_nodes);
}